// EdgeConvBlock_44693429682219
// MI455X (gfx1250) — compile-verified
//
#include <hip/hip_runtime.h>
#include <hip/hip_bf16.h>
#include <math.h>

#define BB 16
#define NN 1024
#define DD 64
#define CC 64
#define KK 20

typedef float v2f __attribute__((ext_vector_type(2)));
typedef float v8f __attribute__((ext_vector_type(8)));
typedef unsigned int u32x4 __attribute__((ext_vector_type(4)));
typedef int i32x4 __attribute__((ext_vector_type(4)));
typedef int i32x8 __attribute__((ext_vector_type(8)));

// CDNA5 fp32 WMMA: D(16x16) = A(16x4) * B(4x16) + C
static __device__ __forceinline__ v8f wmma_f32x4(v2f a, v2f b, v8f c) {
    return __builtin_amdgcn_wmma_f32_16x16x4_f32(false, a, false, b, (short)0, c, false, false);
}

// Tensor Data Mover: 1-row 2D descriptor copying `ndwords` DWORDs from global
// to LDS offset `lds_off_bytes`. D# per CDNA5 ISA ch.8 (group0/group1 bitfields).
static __device__ __forceinline__ void tdm_load_dwords(unsigned int lds_off_bytes,
                                                       const void* g,
                                                       unsigned int ndwords) {
    unsigned long long ga = (unsigned long long)(size_t)g;
    u32x4 g0 = { 1u,                                   // count=1, user mode
                 lds_off_bytes,                        // lds_addr
                 (unsigned int)ga,                     // global_addr[31:0]
                 (unsigned int)((ga >> 32) & 0x01FFFFFFull) | 0x80000000u }; // addr[56:32] | type=2
    i32x8 g1 = { (int)(2u << 16),                      // wg_mask=0, data_size=4B
                 (int)((ndwords & 0xFFFFu) << 16),     // tensor_dim0[15:0] @ bits63:48
                 (int)(((ndwords >> 16) & 0xFFFFu) | (1u << 16)), // tdim0 hi | tensor_dim1=1
                 (int)((ndwords & 0xFFFFu) << 16),     // tile_dim0 @ bits127:112
                 1,                                    // tile_dim1=1, tile_dim2=0
                 (int)ndwords,                         // tensor_dim0_stride[31:0]
                 0, 0 };
    i32x4 gz = { 0, 0, 0, 0 };
#if __clang_major__ >= 23
    i32x8 gz8 = { 0, 0, 0, 0, 0, 0, 0, 0 };
    __builtin_amdgcn_tensor_load_to_lds(g0, g1, gz, gz, gz8, 0);
#else
    __builtin_amdgcn_tensor_load_to_lds(g0, g1, gz, gz, 0);
#endif
}

// ---------------------------------------------------------------- row norms
__global__ void ec_norms(const float* __restrict__ x, float* __restrict__ x2) {
    int r = blockIdx.x * blockDim.x + threadIdx.x;
    if (r >= BB * NN) return;
    const float4* p = (const float4*)(x + (size_t)r * DD);
    float s = 0.f;
#pragma unroll
    for (int i = 0; i < DD / 4; ++i) {
        float4 v = p[i];
        s += v.x * v.x + v.y * v.y + v.z * v.z + v.w * v.w;
    }
    x2[r] = s;
}

// ------------------------------------------------------------ lengths, k_eff
__global__ void ec_keff(const unsigned char* __restrict__ mask,
                        int* __restrict__ lens, int* __restrict__ keff) {
    __shared__ int sl[BB];
    int t = threadIdx.x;
    if (t < BB) {
        int c = 0;
        for (int j = 0; j < NN; ++j) c += (mask[t * NN + j] != 0);
        sl[t] = c;
    }
    __syncthreads();
    if (t == 0) {
        int mn = sl[0];
        for (int b = 1; b < BB; ++b) mn = min(mn, sl[b]);
        int kglob = min(KK, max(1, mn - 1));
        for (int b = 0; b < BB; ++b) {
            lens[b] = sl[b];
            float kp = sqrtf((float)sl[b] / 50.0f) * 8.0f;
            kp = fminf(fmaxf(kp, 8.0f), 20.0f);
            keff[b] = min((int)kp, kglob);
        }
    }
}

// ------------------------------------------------- KNN: dist strip + top-20
// grid: B*64 blocks, 128 threads (4 waves). Wave w computes col-tiles w,w+4,...
// LDS: xtile[16][68] | x2row[16] | dist[16][1024]
__global__ void ec_knn(const float* __restrict__ x, const float* __restrict__ x2,
                       const int* __restrict__ lens, const int* __restrict__ keff,
                       int* __restrict__ idxs) {
    extern __shared__ float smem[];
    float* xtile = smem;                 // 16*68
    float* x2row = xtile + 16 * 68;      // 16
    float* dist  = x2row + 16;           // 16*1024

    const int b     = blockIdx.x >> 6;
    const int rBase = (blockIdx.x & 63) * 16;
    const int tid   = threadIdx.x;
    const int lane  = tid & 31;
    const int wave  = tid >> 5;
    const int len   = lens[b];
    const int ke    = keff[b];

    // stage X row tile [16][64] -> padded stride 68 (float4 path)
    {
        int row = tid >> 3, j0 = (tid & 7) * 8;
        const float4* src = (const float4*)(x + ((size_t)b * NN + rBase + row) * DD + j0);
        float4* dst = (float4*)(xtile + row * 68 + j0);
        dst[0] = src[0];
        dst[1] = src[1];
        if (tid < 16) x2row[tid] = x2[b * NN + rBase + tid];
    }
    __syncthreads();

    // preload 16 A-fragments (K=0..63 in steps of 4)
    const int m    = lane & 15;
    const int hsel = lane >> 4;
    v2f afr[16];
#pragma unroll
    for (int kk = 0; kk < 16; ++kk) {
        int kc = kk * 4 + 2 * hsel;
        afr[kk] = *(const v2f*)(xtile + m * 68 + kc);
    }

    const int n = lane & 15;
    for (int ct = wave; ct < 64; ct += 4) {
        const int cb = ct * 16;
        const float* xcol = x + ((size_t)b * NN + cb + n) * DD;
        if (ct + 4 < 64) __builtin_prefetch(xcol + 64 * DD, 0, 0);   // next strip
        // batch all 16 B-fragment loads; barrier keeps them clustered (clause + 1 wait)
        v2f bfr[16];
#pragma unroll
        for (int kk = 0; kk < 16; ++kk) {
            int kr = kk * 4 + 2 * hsel;
            bfr[kk] = *(const v2f*)(xcol + kr);
        }
        asm volatile("" ::: "memory");   // keep loads grouped ahead of the WMMA chain
        // two independent accumulator chains -> 2-way ILP through the XDL pipe
        v8f acc0 = {0.f, 0.f, 0.f, 0.f, 0.f, 0.f, 0.f, 0.f};
        v8f acc1 = {0.f, 0.f, 0.f, 0.f, 0.f, 0.f, 0.f, 0.f};
#pragma unroll
        for (int kk = 0; kk < 16; kk += 2) {
            acc0 = wmma_f32x4(afr[kk],     bfr[kk],     acc0);
            acc1 = wmma_f32x4(afr[kk + 1], bfr[kk + 1], acc1);
        }
        v8f acc = acc0 + acc1;

        const int   col   = cb + n;
        const float x2col = x2[b * NN + col];
#pragma unroll
        for (int v = 0; v < 8; ++v) {
            int mm = v + 8 * hsel;
            float d = x2row[mm] + x2col - 2.0f * acc[v];
            d = fmaxf(d, 0.0f);
            int gr = rBase + mm;
            if (col == gr || col >= len) d = __builtin_inff();
            dist[mm * NN + col] = d;
        }
    }
    __syncthreads();

    // top-20 per row: 4 rows per wave, argmin sweeps with lowest-index ties
    for (int mrow = wave * 4; mrow < wave * 4 + 4; ++mrow) {
        const int gr = rBase + mrow;
        for (int t = 0; t < KK; ++t) {
            float bv = __builtin_inff();
            int   bi = NN;
#pragma unroll
            for (int j = 0; j < NN / 32; ++j) {
                int   c = lane + 32 * j;
                float v = dist[mrow * NN + c];
                if (v < bv) { bv = v; bi = c; }      // ascending scan keeps lowest index
            }
#pragma unroll
            for (int off = 16; off >= 1; off >>= 1) {
                float ov = __shfl_xor(bv, off, 32);
                int   oi = __shfl_xor(bi, off, 32);
                if (ov < bv || (ov == bv && oi < bi)) { bv = ov; bi = oi; }
            }
            if (lane == 0) {
                dist[mrow * NN + bi] = __builtin_inff();
                idxs[((size_t)b * NN + gr) * KK + t] = (t < ke) ? bi : 0;
            }
        }
    }
}

// ---------------------------------------- fused gather + MLP + GN + ReLU + max
// grid: B*64 blocks, 128 threads (4 waves). Wave w handles kk = w, w+4, ...
// LDS floats: W1s[8192]@0 | W2s[4096]@32768B | ps[256] | xit[16*68] | idxt[320]
//             | A[4*16*132] | H[4*16*66] | O[4*1024]
__global__ void ec_edge(const float* __restrict__ x, const int* __restrict__ idxs,
                        const float* __restrict__ W1, const float* __restrict__ b1,
                        const float* __restrict__ gamma, const float* __restrict__ beta,
                        const float* __restrict__ W2, const float* __restrict__ b2,
                        const int* __restrict__ lens, float* __restrict__ out) {
    extern __shared__ float smem[];
    float* W1s  = smem;                    // 8192  (LDS byte offset 0)
    float* W2s  = W1s + 128 * 64;          // 4096  (LDS byte offset 32768)
    float* ps   = W2s + 64 * 64;           // 256: b1|gamma|beta|b2
    float* xit  = ps + 256;                // 16*68
    int*   idxt = (int*)(xit + 16 * 68);   // 320
    float* Abuf = (float*)(idxt + 320);    // 4*16*132
    float* Hbuf = Abuf + 4 * 16 * 132;     // 4*16*66
    float* Omax = Hbuf + 4 * 16 * 66;      // 4*1024

    const int b     = blockIdx.x >> 6;
    const int pBase = (blockIdx.x & 63) * 16;
    const int tid   = threadIdx.x;
    const int lane  = tid & 31;
    const int wave  = tid >> 5;
    const int len   = lens[b];

    // W1/W2 staged by the Tensor Data Mover (one wave issues; EXEC ignored by TDM)
    if (wave == 0) {
        tdm_load_dwords(0u, W1, 128 * 64);
        tdm_load_dwords(32768u, W2, 64 * 64);
        __builtin_amdgcn_s_wait_tensorcnt(0);
    }
    // params / point tile / idx tile with vector loads
    if (tid < 64)        ps[tid]              = b1[tid];
    else                 ps[64 + (tid - 64)]  = gamma[tid - 64];
    if (tid < 64)        ps[128 + tid]        = beta[tid];
    else                 ps[192 + (tid - 64)] = b2[tid - 64];
    {
        int row = tid >> 3, j0 = (tid & 7) * 8;
        const float4* src = (const float4*)(x + ((size_t)b * NN + pBase + row) * DD + j0);
        float4* dst = (float4*)(xit + row * 68 + j0);
        dst[0] = src[0];
        dst[1] = src[1];
    }
    for (int i = tid; i < 16 * KK; i += 128)
        idxt[i] = idxs[((size_t)b * NN + pBase + (i / KK)) * KK + (i % KK)];
    __syncthreads();

    float* Aw = Abuf + wave * 16 * 132;
    float* Hw = Hbuf + wave * 16 * 66;
    const int m    = lane & 15;
    const int hsel = lane >> 4;
    const int n    = lane & 15;

    v8f vmax[4];
#pragma unroll
    for (int ct = 0; ct < 4; ++ct)
#pragma unroll
        for (int v = 0; v < 8; ++v) vmax[ct][v] = -__builtin_inff();

    for (int kk = wave; kk < KK; kk += 4) {
        // build 16 edge rows [xi | xnb - xi], 2 lanes per row, float4 path
        {
            int row = lane >> 1, half = lane & 1;
            int nb  = idxt[row * KK + kk];
            const float4* xnb4 = (const float4*)(x + ((size_t)b * NN + nb) * DD + half * 32);
            const float*  xir  = xit + row * 68 + half * 32;
            float* ArowA = Aw + row * 132 + half * 32;
            float* ArowB = ArowA + 64;
#pragma unroll
            for (int j4 = 0; j4 < 8; ++j4) {
                float4 vn = xnb4[j4];
                float4 vi = *(const float4*)(xir + j4 * 4);
                *(float4*)(ArowA + j4 * 4) = vi;
                float4 d = { vn.x - vi.x, vn.y - vi.y, vn.z - vi.z, vn.w - vi.w };
                *(float4*)(ArowB + j4 * 4) = d;
            }
        }
        // GEMM1: [16,128] @ W1[128,64]  (4 independent ct chains)
        v8f acc[4];
#pragma unroll
        for (int ct = 0; ct < 4; ++ct) acc[ct] = (v8f){0.f,0.f,0.f,0.f,0.f,0.f,0.f,0.f};
        for (int k0 = 0; k0 < 32; ++k0) {
            int kc = k0 * 4 + 2 * hsel;
            v2f a = *(const v2f*)(Aw + m * 132 + kc);
#pragma unroll
            for (int ct = 0; ct < 4; ++ct) {
                v2f bw;
                bw.x = W1s[kc * 64 + ct * 16 + n];
                bw.y = W1s[(kc + 1) * 64 + ct * 16 + n];
                acc[ct] = wmma_f32x4(a, bw, acc[ct]);
            }
        }
        // +b1, spill H to LDS
#pragma unroll
        for (int ct = 0; ct < 4; ++ct) {
            int col = ct * 16 + n;
#pragma unroll
            for (int v = 0; v < 8; ++v)
                Hw[(v + 8 * hsel) * 66 + col] = acc[ct][v] + ps[col];
        }
        // GroupNorm(16 groups of 4) + ReLU, 2 lanes per row
        {
            int row = lane >> 1, c0 = (lane & 1) * 32;
#pragma unroll
            for (int g = 0; g < 8; ++g) {
                int   c  = c0 + g * 4;
                float h0 = Hw[row * 66 + c],     h1 = Hw[row * 66 + c + 1];
                float h2 = Hw[row * 66 + c + 2], h3 = Hw[row * 66 + c + 3];
                float mu = 0.25f * (h0 + h1 + h2 + h3);
                float sq = 0.25f * (h0 * h0 + h1 * h1 + h2 * h2 + h3 * h3);
                float rs = rsqrtf(fmaxf(sq - mu * mu, 0.0f) + 1e-5f);
                Hw[row * 66 + c]     = fmaxf((h0 - mu) * rs * ps[64 + c]     + ps[128 + c],     0.f);
                Hw[row * 66 + c + 1] = fmaxf((h1 - mu) * rs * ps[64 + c + 1] + ps[128 + c + 1], 0.f);
                Hw[row * 66 + c + 2] = fmaxf((h2 - mu) * rs * ps[64 + c + 2] + ps[128 + c + 2], 0.f);
                Hw[row * 66 + c + 3] = fmaxf((h3 - mu) * rs * ps[64 + c + 3] + ps[128 + c + 3], 0.f);
            }
        }
        // GEMM2: [16,64] @ W2[64,64], +b2, running max
        v8f acc2[4];
#pragma unroll
        for (int ct = 0; ct < 4; ++ct) acc2[ct] = (v8f){0.f,0.f,0.f,0.f,0.f,0.f,0.f,0.f};
        for (int k0 = 0; k0 < 16; ++k0) {
            int kc = k0 * 4 + 2 * hsel;
            v2f a = *(const v2f*)(Hw + m * 66 + kc);
#pragma unroll
            for (int ct = 0; ct < 4; ++ct) {
                v2f bw;
                bw.x = W2s[kc * 64 + ct * 16 + n];
                bw.y = W2s[(kc + 1) * 64 + ct * 16 + n];
                acc2[ct] = wmma_f32x4(a, bw, acc2[ct]);
            }
        }
#pragma unroll
        for (int ct = 0; ct < 4; ++ct) {
            int col = ct * 16 + n;
#pragma unroll
            for (int v = 0; v < 8; ++v)
                vmax[ct][v] = fmaxf(vmax[ct][v], acc2[ct][v] + ps[192 + col]);
        }
    }

    // per-wave max tile -> LDS, then cross-wave reduce + masked store
#pragma unroll
    for (int ct = 0; ct < 4; ++ct)
#pragma unroll
        for (int v = 0; v < 8; ++v)
            Omax[wave * 1024 + (v + 8 * hsel) * 64 + ct * 16 + n] = vmax[ct][v];
    __syncthreads();

    for (int o = tid * 8; o < tid * 8 + 8; ++o) {
        int pt = o >> 6, ch = o & 63;
        float r = Omax[o];
        r = fmaxf(r, Omax[1024 + o]);
        r = fmaxf(r, Omax[2048 + o]);
        r = fmaxf(r, Omax[3072 + o]);
        int gp = pBase + pt;
        out[((size_t)b * NN + gp) * CC + ch] = (gp < len) ? r : 0.0f;
    }
}

// ---------------------------------------------------------------- launcher
extern "C" void kernel_launch(void* const* d_in, const int* in_sizes, int n_in,
                              void* d_out, int out_size, void* d_ws, size_t ws_size,
                              hipStream_t stream) {
    const float*         x     = (const float*)d_in[0];
    const unsigned char* mask  = (const unsigned char*)d_in[1];
    const float*         W1    = (const float*)d_in[2];
    const float*         b1    = (const float*)d_in[3];
    const float*         gamma = (const float*)d_in[4];
    const float*         beta  = (const float*)d_in[5];
    const float*         W2    = (const float*)d_in[6];
    const float*         b2    = (const float*)d_in[7];
    float*               out   = (float*)d_out;

    char* ws   = (char*)d_ws;
    int*  idxs = (int*)ws;                                       // B*N*20 ints
    float* x2  = (float*)(ws + (size_t)BB * NN * KK * 4);        // B*N floats
    int*  lens = (int*)(ws + (size_t)BB * NN * KK * 4 + (size_t)BB * NN * 4);
    int*  keff = lens + BB;

    ec_norms<<<(BB * NN + 255) / 256, 256, 0, stream>>>(x, x2);
    ec_keff<<<1, 32, 0, stream>>>(mask, lens, keff);

    size_t knnShmem  = (size_t)(16 * 68 + 16 + 16 * NN) * sizeof(float);
    ec_knn<<<BB * 64, 128, knnShmem, stream>>>(x, x2, lens, keff, idxs);

    size_t edgeShmem = (size_t)(128 * 64 + 64 * 64 + 256 + 16 * 68 + 320 +
                                4 * 16 * 132 + 4 * 16 * 66 + 4 * 1024) * sizeof(float);
    ec_edge<<<BB * 64, 128, edgeShmem, stream>>>(x, idxs, W1, b1, gamma, beta,
                                                 W2, b2, lens, out);
}